// LinearCRFSemanticRoleLabeling_33904471834765
// MI455X (gfx1250) — compile-verified
//
#include <hip/hip_runtime.h>
#include <stdint.h>

#define SS 512
#define BB 16
#define NCH (BB * SS)          // 8192 chains
#define TPB 64                 // threads per block in crf_main
#define CH 16                  // words per staged chunk
#define NCHUNK (SS / CH)       // 32
#define NEGINF (-INFINITY)
#define TSTRIDE 36             // floats per thread per buffer (16 + 16 + 4 pad) = 144B, 16B aligned

// ws layout (bytes)
#define WS_BP    0u                            // uint16[512 * NCH]  = 8,388,608 B
#define WS_LENS  (512u * NCH * 2u)             // int[16]
#define WS_WMB   (WS_LENS + 64u)               // uint32[16*16] word-mask bitmaps
#define WS_CNT   (WS_WMB + 1024u)              // int count
#define WS_LIST  (WS_CNT + 64u)                // int[NCH] active chain ids

// Low 32 bits of a generic (flat) pointer into LDS == LDS byte offset (ISA 10.2:
// the LDS address is the low 32 bits of the shared-aperture address).
__device__ __forceinline__ unsigned lds_off(const void* p) {
    return (unsigned)(uintptr_t)p;
}

__device__ __forceinline__ void async_b128(unsigned lds, const void* gaddr) {
    asm volatile("global_load_async_to_lds_b128 %0, %1, off"
                 :: "v"(lds), "v"(gaddr) : "memory");
}

// ---- kernel 1: per-batch length, word-mask bitmap, zero the compaction counter
extern "C" __global__ void crf_prep(const int* __restrict__ prd,
                                    int* __restrict__ lens,
                                    uint32_t* __restrict__ wmb,
                                    int* __restrict__ count) {
    int b = threadIdx.x;
    if (b == 0) *count = 0;
    if (b >= BB) return;
    int s = 0;
    for (int k = 0; k < 16; ++k) {
        uint32_t bits = 0;
        for (int j = 0; j < 32; ++j) {
            int v = (prd[b * SS + k * 32 + j] != 0) ? 1 : 0;
            s += v;
            bits |= ((uint32_t)v) << j;
        }
        wmb[b * 16 + k] = bits;
    }
    lens[b] = s;
}

// ---- kernel 2: zero the entire output (reference output is mostly false)
extern "C" __global__ void crf_zero(float4* __restrict__ out4, int n4) {
    int i = blockIdx.x * blockDim.x + threadIdx.x;
    if (i < n4) out4[i] = make_float4(0.f, 0.f, 0.f, 0.f);
}

// ---- kernel 3: compact active chains (gold & prd_mask) into a dense list
extern "C" __global__ void crf_compact(const int* __restrict__ prd,
                                       const int* __restrict__ gold,
                                       int* __restrict__ count,
                                       int* __restrict__ list) {
    int n = blockIdx.x * blockDim.x + threadIdx.x;
    if (n >= NCH) return;
    if ((gold[n] != 0) && (prd[n] != 0)) {      // prd_mask[b, p] == prd[n] for n = b*S+p
        int idx = atomicAdd(count, 1);
        list[idx] = n;
    }
}

// ---- kernel 4: dense Viterbi over active chains with async double-buffered LDS staging
extern "C" __global__ void __launch_bounds__(TPB)
crf_main(const float* __restrict__ sbg, const float* __restrict__ seg,
         const int* __restrict__ abm, const int* __restrict__ aem,
         const int* __restrict__ lens, const uint32_t* __restrict__ wmb,
         const int* __restrict__ count, const int* __restrict__ list,
         uint16_t* __restrict__ bp, float* __restrict__ out)
{
    __shared__ __align__(16) float stage[2 * TPB * TSTRIDE];  // 18 KB double buffer
    __shared__ uint32_t wmb_sh[BB * 16];                      // all batches' word-mask bitmaps

    const int t = threadIdx.x;
    for (int i = t; i < BB * 16; i += TPB) wmb_sh[i] = wmb[i];
    __syncthreads();

    const int ci = blockIdx.x * TPB + t;        // compacted chain index
    if (ci >= *count) return;
    const int n = list[ci];
    const int b = n >> 9;
    const int p = n & (SS - 1);

    const size_t rbase = ((size_t)b * SS + p) * SS;
    const float* rowb = sbg + rbase;
    const float* rowe = seg + rbase;
    const int*   rowbm = abm + rbase;
    const int*   rowem = aem + rbase;
    const int lenb = lens[b];
    float* outrow = out + (size_t)n * (SS * 2);

    const unsigned lbase0 = lds_off(&stage[t * TSTRIDE]);
    const unsigned lbase1 = lds_off(&stage[(TPB + t) * TSTRIDE]);

    auto issue = [&](int c, unsigned lbase) {
        const char* gb = (const char*)(rowb + c * CH);
        const char* ge = (const char*)(rowe + c * CH);
#pragma unroll
        for (int k = 0; k < 4; ++k) {
            async_b128(lbase + k * 16, gb + k * 16);            // begin-scores row
            async_b128(lbase + 64 + k * 16, ge + k * 16);       // end-scores row
        }
    };

    float o0 = NEGINF, o1 = NEGINF, o2 = NEGINF, o3 = NEGINF, o4v = NEGINF;

    issue(0, lbase0);
    for (int c = 0; c < NCHUNK; ++c) {
        const float* sm = &stage[((c & 1) ? (TPB + t) : t) * TSTRIDE];
        if (c + 1 < NCHUNK) {
            issue(c + 1, (c & 1) ? lbase0 : lbase1);
            asm volatile("s_wait_asynccnt 8" ::: "memory");     // chunk c landed (in-order)
        } else {
            asm volatile("s_wait_asynccnt 0" ::: "memory");
        }

        // this chunk's 16 word-mask bits (one ds_load per chunk)
        const uint32_t cb = wmb_sh[b * 16 + (c >> 1)];
        const uint32_t bits16 = cb >> ((c & 1) * 16);

        // preload this chunk's mask words into registers
        int4 mb4[CH / 4], me4[CH / 4];
        const int4* pb4 = (const int4*)(rowbm + c * CH);
        const int4* pe4 = (const int4*)(rowem + c * CH);
#pragma unroll
        for (int k = 0; k < CH / 4; ++k) { mb4[k] = pb4[k]; me4[k] = pe4[k]; }
        const int* mbv = (const int*)mb4;
        const int* mev = (const int*)me4;

#pragma unroll
        for (int j = 0; j < CH; ++j) {
            const int w = c * CH + j;
            if (w == 0) continue;                 // word 0 (ROOT) excluded from the chain
            const float sb = sm[j];
            const float se = sm[16 + j];
            float pab = 1.f / (1.f + expf(-sb)); if (mbv[j] == 0) pab = 0.f;
            float pae = 1.f / (1.f + expf(-se)); if (mev[j] == 0) pae = 0.f;
            const float qb = 1.f - pab, qe = 1.f - pae;
            const float eB = logf(pab * qe);
            const float eE = logf(qb * pae);
            const float eS = logf(pab * pae);
            const float eO = logf(qb * qe);
            const float eI = (w == p) ? NEGINF : eO;   // eye() kills tag I on the diagonal

            if (w == 1) {
                // opt0 = emit + start_mask  (start allows B,S,O)
                o0 = eB; o1 = NEGINF; o2 = NEGINF; o3 = eS; o4v = eO;
            } else {
                const int wm = (int)((bits16 >> j) & 1u);
                // allowed prev sets: cur in {B,S,O} <- {E,S,O}; cur in {I,E} <- {B,I}
                float b234 = NEGINF; int i234 = 0;     // argmax ties -> lowest prev index
                if (o2  > b234) { b234 = o2;  i234 = 2; }
                if (o3  > b234) { b234 = o3;  i234 = 3; }
                if (o4v > b234) { b234 = o4v; i234 = 4; }
                float b01 = o0; int i01 = 0;           // prev 0 allowed -> init matches argmax
                if (o1 > b01) { b01 = o1; i01 = 1; }
                uint32_t packed = 0;
                if (wm) {
                    const float n1 = b01 + eI;
                    const float n2 = b01 + eE;
                    o0 = b234 + eB;
                    o3 = b234 + eS;
                    o4v = b234 + eO;
                    o1 = n1; o2 = n2;
                    packed = (uint32_t)i234 | ((uint32_t)i01 << 3) | ((uint32_t)i01 << 6)
                           | ((uint32_t)i234 << 9) | ((uint32_t)i234 << 12);
                }
                bp[(size_t)(w - 1) * NCH + ci] = (uint16_t)packed;   // coalesced across lanes
            }
        }
    }

    // final = opt + end_mask (allows E,S,O); first-occurrence argmax
    float fb = NEGINF; int fi = 0;
    if (o2  > fb) { fb = o2;  fi = 2; }
    if (o3  > fb) { fb = o3;  fi = 3; }
    if (o4v > fb) { fb = o4v; fi = 4; }
    bp[(size_t)(SS - 1) * NCH + ci] = 0;                // position 511 never written by loop
    bp[(size_t)lenb * NCH + ci] = (uint16_t)fi;         // backpoints[n, lens, 0] = argmax

    // backtrace: i = 511 .. 1, tag of word i = bp[i][pt]; output pre-zeroed -> store 1.0 only
    int pt = 0;
    for (int i = SS - 1; i >= 1; --i) {
        const uint32_t pk = bp[(size_t)i * NCH + ci];
        pt = (int)((pk >> (3 * pt)) & 7u);
        const bool tb = (pt == 0) | (pt == 3);          // B or S
        const bool te = (pt == 2) | (pt == 3);          // E or S
        if (tb && (rowbm[i] != 0)) outrow[2 * i]     = 1.f;
        if (te && (rowem[i] != 0)) outrow[2 * i + 1] = 1.f;
    }
}

extern "C" void kernel_launch(void* const* d_in, const int* in_sizes, int n_in,
                              void* d_out, int out_size, void* d_ws, size_t ws_size,
                              hipStream_t stream) {
    const float* sbg = (const float*)d_in[0];   // s_arg_begin [16,512,512] f32
    const float* seg = (const float*)d_in[1];   // s_arg_end   [16,512,512] f32
    const int* prd   = (const int*)d_in[2];     // prd_mask [16,512]
    const int* abm   = (const int*)d_in[3];     // arg_begin_mask [16,512,512]
    const int* aem   = (const int*)d_in[4];     // arg_end_mask   [16,512,512]
    const int* gold  = (const int*)d_in[5];     // gold_prds [16,512]

    char* ws = (char*)d_ws;
    uint16_t* bp   = (uint16_t*)(ws + WS_BP);
    int*      lens = (int*)(ws + WS_LENS);
    uint32_t* wmb  = (uint32_t*)(ws + WS_WMB);
    int*      cnt  = (int*)(ws + WS_CNT);
    int*      list = (int*)(ws + WS_LIST);
    float*    out  = (float*)d_out;             // [16,512,512,2] as 0.0/1.0

    const int n4 = NCH * SS * 2 / 4;            // output float4 count
    crf_prep<<<1, 32, 0, stream>>>(prd, lens, wmb, cnt);
    crf_zero<<<(n4 + 255) / 256, 256, 0, stream>>>((float4*)out, n4);
    crf_compact<<<NCH / 256, 256, 0, stream>>>(prd, gold, cnt, list);
    crf_main<<<NCH / TPB, TPB, 0, stream>>>(sbg, seg, abm, aem, lens, wmb, cnt, list, bp, out);
}